// GCN_23553600651661
// MI455X (gfx1250) — compile-verified
//
#include <hip/hip_runtime.h>
#include <stdint.h>

// ---------------------------------------------------------------------------
// GCN on MI455X (gfx1250). Dense transforms use V_WMMA_F32_16X16X4_F32
// (exact fp32, matches the fp32 reference). W matrices are pre-transposed
// (and the 40-wide output layer padded to 48) so the WMMA inner loop has
// zero divergence, contiguous b64 loads, and compile-time strides.
// Each wave processes 5 M-subtiles sharing one B fragment per K-step:
// 5 independent back-to-back WMMAs, 5x B-load amortization.
// Aggregation: block-per-edge scatter (uniform scalar index loads) with f32
// global atomics resident in the 192MB L2 (accumulators are 51MB).
// ---------------------------------------------------------------------------

typedef float v2f __attribute__((ext_vector_type(2)));
typedef float v8f __attribute__((ext_vector_type(8)));

#define GCN_NODES 100000
#define GCN_EDGES 1600000
#define GCN_INCH  500
#define GCN_HID   128
#define GCN_OUT   40
#define GCN_OUTP  48     // padded to 3 N-tiles

// ---------------------------- degree / norm --------------------------------

__global__ void k_init_deg(float* __restrict__ deg, int n) {
  int i = blockIdx.x * blockDim.x + threadIdx.x;
  if (i < n) deg[i] = 1.0f;  // self-loop weight
}

__global__ void k_accum_deg(const long long* __restrict__ dst,
                            const float* __restrict__ ew,
                            float* __restrict__ deg, int E) {
  int e = blockIdx.x * blockDim.x + threadIdx.x;
  if (e < E) atomicAdd(&deg[(int)dst[e]], ew[e]);
}

__global__ void k_dinv(float* __restrict__ deg, int n) {
  int i = blockIdx.x * blockDim.x + threadIdx.x;
  if (i < n) {
    float d = deg[i];
    deg[i] = (d > 0.0f) ? rsqrtf(d) : 0.0f;  // becomes dinv in place
  }
}

// ------------------- weight transpose (+ column pad) -----------------------
// Wt[n*K + k] = (n < Nin) ? W[k*Nin + n] : 0    for n in [0,Npad), k in [0,K)
__global__ void k_transpose_pad(const float* __restrict__ W, float* __restrict__ Wt,
                                int K, int Nin, int Npad) {
  int idx = blockIdx.x * blockDim.x + threadIdx.x;
  if (idx < Npad * K) {
    int n = idx / K;
    int k = idx - n * K;
    Wt[idx] = (n < Nin) ? W[(size_t)k * Nin + n] : 0.0f;
  }
}

// ------------------------------- WMMA GEMM ---------------------------------
// C[M x NSTORE] = A[M x K] @ Wt^T, Wt is column-major [N x K], N%16==0.
// Grid: M/(16*MSUB) blocks of 32*(N/16) threads. Wave w owns columns
// [16w,16w+16); each wave accumulates MSUB 16-row subtiles.

__device__ inline v8f wmma_f32x4(v2f a, v2f b, v8f c) {
  return __builtin_amdgcn_wmma_f32_16x16x4_f32(
      /*neg_a=*/false, a, /*neg_b=*/false, b,
      /*c_mod=*/(short)0, c, /*reuse_a=*/false, /*reuse_b=*/false);
}

template <int K, int N, int NSTORE, int MSUB>
__global__ void k_gemm_wmma(const float* __restrict__ A,
                            const float* __restrict__ Wt,
                            float* __restrict__ C) {
  const int lane = threadIdx.x & 31;
  const int wave = threadIdx.x >> 5;
  const int half = lane >> 4;            // 0: lanes 0-15, 1: lanes 16-31
  const int l16  = lane & 15;

  const int mBase = blockIdx.x * (16 * MSUB);
  const int nc    = wave * 16 + l16;     // always < N (padded)

  // per-lane walking pointers; strides are compile-time constants
  const float* wPtr = Wt + (size_t)nc * K + 2 * half;
  const float* aPtr[MSUB];
#pragma unroll
  for (int m = 0; m < MSUB; ++m)
    aPtr[m] = A + (size_t)(mBase + 16 * m + l16) * K + 2 * half;

  v8f acc[MSUB];
#pragma unroll
  for (int m = 0; m < MSUB; ++m)
    acc[m] = (v8f){0.f, 0.f, 0.f, 0.f, 0.f, 0.f, 0.f, 0.f};

#pragma unroll 2
  for (int k = 0; k < K; k += 4) {
    v2f b;
    b.x = wPtr[0];
    b.y = wPtr[1];
    wPtr += 4;
#pragma unroll
    for (int m = 0; m < MSUB; ++m) {
      v2f a;
      a.x = aPtr[m][0];
      a.y = aPtr[m][1];
      aPtr[m] += 4;
      acc[m] = wmma_f32x4(a, b, acc[m]);   // MSUB independent WMMAs per step
    }
  }

  if (N == NSTORE || nc < NSTORE) {
#pragma unroll
    for (int m = 0; m < MSUB; ++m) {
      const int row0 = mBase + 16 * m + half * 8;
#pragma unroll
      for (int g = 0; g < 8; ++g)
        C[(size_t)(row0 + g) * NSTORE + nc] = acc[m][g];
    }
  }
}

// --------------------------- aggregation -----------------------------------
// out[i,:] = H[i,:] * dinv[i]^2   (self-loop term, atomic-free init)
__global__ void k_self_init(const float* __restrict__ H,
                            const float* __restrict__ dinv,
                            float* __restrict__ out,
                            long long total, int F) {
  long long t = (long long)blockIdx.x * blockDim.x + threadIdx.x;
  if (t < total) {
    int i = (int)(t / F);
    float d = dinv[i];
    out[t] = H[t] * d * d;
  }
}

// One block per edge: src/dst/ew become workgroup-uniform scalar loads.
__global__ void k_scatter(const float* __restrict__ H,
                          const long long* __restrict__ src,
                          const long long* __restrict__ dst,
                          const float* __restrict__ ew,
                          const float* __restrict__ dinv,
                          float* __restrict__ out, int F) {
  const int e = blockIdx.x;
  const int f = threadIdx.x;
  const int s = (int)src[e];
  const int d = (int)dst[e];
  const float coef = dinv[s] * ew[e] * dinv[d];
  if (f < F)
    atomicAdd(&out[(size_t)d * F + f], H[(size_t)s * F + f] * coef);
}

// ------------------------------ epilogues ----------------------------------

__global__ void k_bias_relu(float* __restrict__ out, const float* __restrict__ b,
                            long long total, int F) {
  long long t = (long long)blockIdx.x * blockDim.x + threadIdx.x;
  if (t < total) {
    float v = out[t] + b[(int)(t % F)];
    out[t] = v > 0.0f ? v : 0.0f;
  }
}

__global__ void k_bias_relu_res(float* __restrict__ out, const float* __restrict__ b,
                                const float* __restrict__ res,
                                long long total, int F) {
  long long t = (long long)blockIdx.x * blockDim.x + threadIdx.x;
  if (t < total) {
    float v = out[t] + b[(int)(t % F)];
    v = v > 0.0f ? v : 0.0f;
    out[t] = v + res[t];
  }
}

__global__ void k_bias(float* __restrict__ out, const float* __restrict__ b,
                       long long total, int F) {
  long long t = (long long)blockIdx.x * blockDim.x + threadIdx.x;
  if (t < total) out[t] = out[t] + b[(int)(t % F)];
}

// In-place log_softmax per row (F=40): one thread per row, tiny traffic.
__global__ void k_logsoftmax(float* __restrict__ out, int n, int F) {
  int i = blockIdx.x * blockDim.x + threadIdx.x;
  if (i >= n) return;
  float* row = out + (size_t)i * F;
  float m = row[0];
  for (int f = 1; f < F; ++f) m = fmaxf(m, row[f]);
  float s = 0.0f;
  for (int f = 0; f < F; ++f) s += __expf(row[f] - m);
  float l = __logf(s);
  for (int f = 0; f < F; ++f) row[f] = row[f] - m - l;
}

// ------------------------------- driver ------------------------------------

static inline size_t align_up(size_t x, size_t a) { return (x + a - 1) & ~(a - 1); }

extern "C" void kernel_launch(void* const* d_in, const int* in_sizes, int n_in,
                              void* d_out, int out_size, void* d_ws, size_t ws_size,
                              hipStream_t stream) {
  (void)in_sizes; (void)n_in; (void)out_size; (void)ws_size;

  const float*     x   = (const float*)d_in[0];
  const long long* ei  = (const long long*)d_in[1];       // [2, E] int64
  const float*     ew  = (const float*)d_in[2];
  const float*     W1  = (const float*)d_in[3];
  const float*     b1  = (const float*)d_in[4];
  const float*     W2  = (const float*)d_in[5];
  const float*     b2  = (const float*)d_in[6];
  const float*     W3  = (const float*)d_in[7];
  const float*     b3  = (const float*)d_in[8];
  float*           out = (float*)d_out;

  const long long* src = ei;
  const long long* dst = ei + GCN_EDGES;

  // workspace layout
  char* ws = (char*)d_ws;
  size_t off = 0;
  float* dinv = (float*)(ws + off); off = align_up(off + sizeof(float) * GCN_NODES, 256);
  float* H    = (float*)(ws + off); off = align_up(off + sizeof(float) * (size_t)GCN_NODES * GCN_HID, 256);
  float* A1   = (float*)(ws + off); off = align_up(off + sizeof(float) * (size_t)GCN_NODES * GCN_HID, 256);
  float* A2   = (float*)(ws + off); off = align_up(off + sizeof(float) * (size_t)GCN_NODES * GCN_HID, 256);
  float* Wt1  = (float*)(ws + off); off = align_up(off + sizeof(float) * GCN_HID * GCN_INCH, 256);
  float* Wt2  = (float*)(ws + off); off = align_up(off + sizeof(float) * GCN_HID * GCN_HID, 256);
  float* Wt3  = (float*)(ws + off); off = align_up(off + sizeof(float) * GCN_OUTP * GCN_HID, 256);

  const long long totH = (long long)GCN_NODES * GCN_HID;   // 12.8M
  const long long totO = (long long)GCN_NODES * GCN_OUT;   // 4M

  // --- gcn_norm: degree with self-loops -> dinv ----------------------------
  k_init_deg<<<(GCN_NODES + 255) / 256, 256, 0, stream>>>(dinv, GCN_NODES);
  k_accum_deg<<<(GCN_EDGES + 255) / 256, 256, 0, stream>>>(dst, ew, dinv, GCN_EDGES);
  k_dinv<<<(GCN_NODES + 255) / 256, 256, 0, stream>>>(dinv, GCN_NODES);

  // --- weight transposes (+ pad W3 to 48 cols) -----------------------------
  k_transpose_pad<<<(GCN_HID * GCN_INCH + 255) / 256, 256, 0, stream>>>(W1, Wt1, GCN_INCH, GCN_HID, GCN_HID);
  k_transpose_pad<<<(GCN_HID * GCN_HID + 255) / 256, 256, 0, stream>>>(W2, Wt2, GCN_HID, GCN_HID, GCN_HID);
  k_transpose_pad<<<(GCN_OUTP * GCN_HID + 255) / 256, 256, 0, stream>>>(W3, Wt3, GCN_HID, GCN_OUT, GCN_OUTP);

  // --- layer 1: h = relu(agg(x@W1) + b1) -----------------------------------
  k_gemm_wmma<GCN_INCH, GCN_HID, GCN_HID, 5>
      <<<GCN_NODES / 80, 32 * (GCN_HID / 16), 0, stream>>>(x, Wt1, H);
  k_self_init<<<(unsigned)((totH + 255) / 256), 256, 0, stream>>>(H, dinv, A1, totH, GCN_HID);
  k_scatter<<<GCN_EDGES, GCN_HID, 0, stream>>>(H, src, dst, ew, dinv, A1, GCN_HID);
  k_bias_relu<<<(unsigned)((totH + 255) / 256), 256, 0, stream>>>(A1, b1, totH, GCN_HID);

  // --- layer 2: h = relu(agg(h@W2) + b2) + h_res ---------------------------
  k_gemm_wmma<GCN_HID, GCN_HID, GCN_HID, 5>
      <<<GCN_NODES / 80, 32 * (GCN_HID / 16), 0, stream>>>(A1, Wt2, H);
  k_self_init<<<(unsigned)((totH + 255) / 256), 256, 0, stream>>>(H, dinv, A2, totH, GCN_HID);
  k_scatter<<<GCN_EDGES, GCN_HID, 0, stream>>>(H, src, dst, ew, dinv, A2, GCN_HID);
  k_bias_relu_res<<<(unsigned)((totH + 255) / 256), 256, 0, stream>>>(A2, b2, A1, totH, GCN_HID);

  // --- layer 3: out = agg(h@W3) + b3, then log_softmax ---------------------
  k_gemm_wmma<GCN_HID, GCN_OUTP, GCN_OUT, 5>
      <<<GCN_NODES / 80, 32 * (GCN_OUTP / 16), 0, stream>>>(A2, Wt3, H);
  k_self_init<<<(unsigned)((totO + 255) / 256), 256, 0, stream>>>(H, dinv, out, totO, GCN_OUT);
  k_scatter<<<GCN_EDGES, 64, 0, stream>>>(H, src, dst, ew, dinv, out, GCN_OUT);
  k_bias<<<(unsigned)((totO + 255) / 256), 256, 0, stream>>>(out, b3, totO, GCN_OUT);
  k_logsoftmax<<<(GCN_NODES + 255) / 256, 256, 0, stream>>>(out, GCN_NODES, GCN_OUT);
}